// mat_GRU_cell_47287589929287
// MI455X (gfx1250) — compile-verified
//
#include <hip/hip_runtime.h>
#include <math.h>

#define ROWS 256
#define COLS 1024
#define LDK  260   // LDS row stride (floats): 260 % 64 = 4 banks apart, keeps 16B alignment

typedef __attribute__((ext_vector_type(2))) float v2f;
typedef __attribute__((ext_vector_type(8))) float v8f;

// ---- order-preserving float <-> uint key (larger float => larger key) ----
__device__ __forceinline__ unsigned f2key(float f) {
  unsigned u = __float_as_uint(f);
  return (u & 0x80000000u) ? ~u : (u | 0x80000000u);
}
__device__ __forceinline__ float key2f(unsigned k) {
  unsigned u = (k & 0x80000000u) ? (k & 0x7FFFFFFFu) : ~k;
  return __uint_as_float(u);
}

// ctl layout (uint words): [0]=prefix [1]=need [2]=cand count [3]=inv_norm bits [4..259]=hist
__global__ void k_init(const float* __restrict__ scorer, unsigned* __restrict__ ctl) {
  __shared__ float red[256];
  int t = threadIdx.x;
  float v = scorer[t];
  red[t] = v * v;
  __syncthreads();
  for (int s = 128; s > 0; s >>= 1) { if (t < s) red[t] += red[t + s]; __syncthreads(); }
  if (t == 0) {
    ctl[0] = 0u; ctl[1] = (unsigned)COLS; ctl[2] = 0u;
    ((float*)ctl)[3] = 1.0f / sqrtf(red[0]);
  }
  ctl[4 + t] = 0u;
}

// scores: one wave per row; 512MB stream of prev_Z -> roofline-dominant kernel
__global__ void k_scores(const float* __restrict__ Z, const float* __restrict__ mask,
                         const float* __restrict__ scorer, const unsigned* __restrict__ ctl,
                         unsigned* __restrict__ keys, int N) {
  const int lane = threadIdx.x & 31;
  const int wave = (int)((blockIdx.x * blockDim.x + threadIdx.x) >> 5);
  const int nw   = (int)((gridDim.x * blockDim.x) >> 5);
  const float4* scv = (const float4*)scorer;
  float4 s0 = scv[2 * lane], s1 = scv[2 * lane + 1];
  float invn = ((const float*)ctl)[3];
  for (int row = wave; row < N; row += nw) {
    const float4* zp = (const float4*)(Z + (size_t)row * ROWS);
    if (row + nw < N)
      __builtin_prefetch(Z + (size_t)(row + nw) * ROWS + lane * 8, 0, 0);
    float4 z0 = zp[2 * lane], z1 = zp[2 * lane + 1];
    float d = z0.x*s0.x + z0.y*s0.y + z0.z*s0.z + z0.w*s0.w
            + z1.x*s1.x + z1.y*s1.y + z1.z*s1.z + z1.w*s1.w;
    #pragma unroll
    for (int off = 16; off; off >>= 1) d += __shfl_xor(d, off, 32);
    if (lane == 0) keys[row] = f2key(d * invn + mask[row]);
  }
}

// radix-select pass: histogram of byte `pass` among keys matching resolved prefix
__global__ void k_hist(const unsigned* __restrict__ keys, unsigned* __restrict__ ctl,
                       int N, int pass) {
  __shared__ unsigned h[256];
  int t = threadIdx.x;
  h[t] = 0u; __syncthreads();
  unsigned prefix = ctl[0];
  int shift = 24 - 8 * pass;
  unsigned hmask = pass ? (0xFFFFFFFFu << (32 - 8 * pass)) : 0u;
  for (int i = (int)(blockIdx.x * blockDim.x) + t; i < N; i += (int)(gridDim.x * blockDim.x)) {
    unsigned k = keys[i];
    if ((k & hmask) == prefix) atomicAdd(&h[(k >> shift) & 255u], 1u);
  }
  __syncthreads();
  if (h[t]) atomicAdd(&ctl[4 + t], h[t]);
}

__global__ void k_select(unsigned* __restrict__ ctl, int pass) {
  int shift = 24 - 8 * pass;
  unsigned need = ctl[1], cum = 0u;
  for (int b = 255; b >= 0; --b) {
    unsigned c = ctl[4 + b];
    cum += c;
    if (cum >= need) { ctl[0] |= ((unsigned)b) << shift; ctl[1] = need - (cum - c); break; }
  }
  for (int i = 0; i < 256; ++i) ctl[4 + i] = 0u;
}

__global__ void k_compact(const unsigned* __restrict__ keys, unsigned* __restrict__ ctl,
                          unsigned* __restrict__ candK, unsigned* __restrict__ candI, int N) {
  unsigned T = ctl[0];
  for (int i = (int)(blockIdx.x * blockDim.x + threadIdx.x); i < N;
       i += (int)(gridDim.x * blockDim.x)) {
    unsigned k = keys[i];
    if (k >= T) {
      unsigned pos = atomicAdd(&ctl[2], 1u);
      if (pos < 2048u) { candK[pos] = k; candI[pos] = (unsigned)i; }
    }
  }
}

// bitonic sort 2048 candidates descending by (key, then index ascending); take top 1024
__global__ __launch_bounds__(1024) void k_sortk(const unsigned* __restrict__ candK,
                                                const unsigned* __restrict__ candI,
                                                const unsigned* __restrict__ ctl,
                                                float* __restrict__ tval,
                                                unsigned* __restrict__ topIdx) {
  __shared__ unsigned long long a[2048];
  int t = threadIdx.x;
  unsigned cnt = ctl[2]; if (cnt > 2048u) cnt = 2048u;
  for (int i = t; i < 2048; i += 1024) {
    unsigned long long v = 0ull;
    if (i < (int)cnt) v = ((unsigned long long)candK[i] << 32) | (unsigned)(~candI[i]);
    a[i] = v;
  }
  __syncthreads();
  for (int k = 2; k <= 2048; k <<= 1) {
    for (int j = k >> 1; j > 0; j >>= 1) {
      #pragma unroll
      for (int rep = 0; rep < 2; ++rep) {
        int i = t + rep * 1024;
        int l = i ^ j;
        if (l > i) {
          unsigned long long x = a[i], y = a[l];
          bool up = (i & k) == 0;                // descending overall
          if (up ? (x < y) : (x > y)) { a[i] = y; a[l] = x; }
        }
      }
      __syncthreads();
    }
  }
  {
    unsigned long long v = a[t];
    unsigned key = (unsigned)(v >> 32);
    topIdx[t] = ~((unsigned)v);
    tval[t]   = tanhf(key2f(key));
  }
}

// Zt[j][k] = prev_Z[idx[j]][k] * tanh(val[j])  (node-major == K-major for WMMA B panels)
__global__ void k_buildz(const float* __restrict__ Z, const float* __restrict__ tval,
                         const unsigned* __restrict__ topIdx, float* __restrict__ Zt) {
  int j = blockIdx.x, t = threadIdx.x;  // 64 threads, float4 each
  unsigned row = topIdx[j];
  float s = tval[j];
  float4 v = ((const float4*)(Z + (size_t)row * ROWS))[t];
  v.x *= s; v.y *= s; v.z *= s; v.w *= s;
  ((float4*)(Zt + (size_t)j * ROWS))[t] = v;
}

// ---- WMMA fragment helpers (V_WMMA_F32_16X16X4_F32, wave32) ----
// A 16x4: lane=M, half selects {K0,K1} vs {K2,K3}   -> float2 at (m, k0+2*half)
// B 4x16: lane&15=N, VGPR=row-in-half, half=+2 rows -> {B[k0+2h][n], B[k0+2h+1][n]}
// C 16x16: M = vgpr + 8*half, N = lane&15

__global__ __launch_bounds__(512) void k_gates_ur(
    const float* __restrict__ Wu, const float* __restrict__ Uu, const float* __restrict__ bu,
    const float* __restrict__ Wr, const float* __restrict__ Ur, const float* __restrict__ br,
    const float* __restrict__ Q,  const float* __restrict__ Zt,
    float* __restrict__ upd, float* __restrict__ RQ) {
  __shared__ float ldsZ[16 * LDK];
  __shared__ float ldsQ[16 * LDK];
  const int tid = threadIdx.x;
  const int n0 = (int)blockIdx.x * 16;
  for (int q = tid; q < 16 * 64; q += 512) {               // Zt panel: [n][k]
    int n = q >> 6, k4 = q & 63;
    float4 v = ((const float4*)(Zt + (size_t)(n0 + n) * ROWS))[k4];
    *(float4*)&ldsZ[n * LDK + 4 * k4] = v;
  }
  for (int e = tid; e < 16 * 256; e += 512) {              // Q panel transposed: [n][k]
    int n = e & 15, k = e >> 4;
    ldsQ[n * LDK + k] = Q[(size_t)k * COLS + n0 + n];
  }
  __syncthreads();

  const int wv = tid >> 5, lane = tid & 31;
  const int half = lane >> 4, l16 = lane & 15;
  const int m0 = wv * 16;
  const float* pWu = Wu + (size_t)(m0 + l16) * ROWS;
  const float* pUu = Uu + (size_t)(m0 + l16) * ROWS;
  const float* pWr = Wr + (size_t)(m0 + l16) * ROWS;
  const float* pUr = Ur + (size_t)(m0 + l16) * ROWS;
  const float* bz = &ldsZ[l16 * LDK];
  const float* bq = &ldsQ[l16 * LDK];

  v8f accU, accR;
  #pragma unroll
  for (int vr = 0; vr < 8; ++vr) {
    size_t o = (size_t)(m0 + vr + 8 * half) * COLS + n0 + l16;
    accU[vr] = bu[o];
    accR[vr] = br[o];
  }
  for (int k0 = 0; k0 < ROWS; k0 += 4) {
    v2f aWu = *(const v2f*)(pWu + k0 + 2 * half);
    v2f aUu = *(const v2f*)(pUu + k0 + 2 * half);
    v2f aWr = *(const v2f*)(pWr + k0 + 2 * half);
    v2f aUr = *(const v2f*)(pUr + k0 + 2 * half);
    v2f bZ  = *(const v2f*)(bz + k0 + 2 * half);
    v2f bQ  = *(const v2f*)(bq + k0 + 2 * half);
    accU = __builtin_amdgcn_wmma_f32_16x16x4_f32(false, aWu, false, bZ, (short)0, accU, false, false);
    accU = __builtin_amdgcn_wmma_f32_16x16x4_f32(false, aUu, false, bQ, (short)0, accU, false, false);
    accR = __builtin_amdgcn_wmma_f32_16x16x4_f32(false, aWr, false, bZ, (short)0, accR, false, false);
    accR = __builtin_amdgcn_wmma_f32_16x16x4_f32(false, aUr, false, bQ, (short)0, accR, false, false);
  }
  #pragma unroll
  for (int vr = 0; vr < 8; ++vr) {
    size_t o = (size_t)(m0 + vr + 8 * half) * COLS + n0 + l16;
    float u = 1.0f / (1.0f + __expf(-accU[vr]));
    float r = 1.0f / (1.0f + __expf(-accR[vr]));
    upd[o] = u;
    RQ[o]  = r * Q[o];
  }
}

__global__ __launch_bounds__(512) void k_gate_h(
    const float* __restrict__ Wh, const float* __restrict__ Uh, const float* __restrict__ bh,
    const float* __restrict__ Q,  const float* __restrict__ Zt, const float* __restrict__ RQ,
    const float* __restrict__ upd, float* __restrict__ out) {
  __shared__ float ldsZ[16 * LDK];
  __shared__ float ldsR[16 * LDK];
  const int tid = threadIdx.x;
  const int n0 = (int)blockIdx.x * 16;
  for (int q = tid; q < 16 * 64; q += 512) {
    int n = q >> 6, k4 = q & 63;
    float4 v = ((const float4*)(Zt + (size_t)(n0 + n) * ROWS))[k4];
    *(float4*)&ldsZ[n * LDK + 4 * k4] = v;
  }
  for (int e = tid; e < 16 * 256; e += 512) {
    int n = e & 15, k = e >> 4;
    ldsR[n * LDK + k] = RQ[(size_t)k * COLS + n0 + n];
  }
  __syncthreads();

  const int wv = tid >> 5, lane = tid & 31;
  const int half = lane >> 4, l16 = lane & 15;
  const int m0 = wv * 16;
  const float* pWh = Wh + (size_t)(m0 + l16) * ROWS;
  const float* pUh = Uh + (size_t)(m0 + l16) * ROWS;
  const float* bz = &ldsZ[l16 * LDK];
  const float* br = &ldsR[l16 * LDK];

  v8f acc;
  #pragma unroll
  for (int vr = 0; vr < 8; ++vr)
    acc[vr] = bh[(size_t)(m0 + vr + 8 * half) * COLS + n0 + l16];
  for (int k0 = 0; k0 < ROWS; k0 += 4) {
    v2f aW = *(const v2f*)(pWh + k0 + 2 * half);
    v2f aU = *(const v2f*)(pUh + k0 + 2 * half);
    v2f bZ = *(const v2f*)(bz + k0 + 2 * half);
    v2f bR = *(const v2f*)(br + k0 + 2 * half);
    acc = __builtin_amdgcn_wmma_f32_16x16x4_f32(false, aW, false, bZ, (short)0, acc, false, false);
    acc = __builtin_amdgcn_wmma_f32_16x16x4_f32(false, aU, false, bR, (short)0, acc, false, false);
  }
  #pragma unroll
  for (int vr = 0; vr < 8; ++vr) {
    size_t o = (size_t)(m0 + vr + 8 * half) * COLS + n0 + l16;
    float h = fmaxf(acc[vr], 0.0f);
    float u = upd[o];
    out[o] = (1.0f - u) * Q[o] + u * h;
  }
}

extern "C" void kernel_launch(void* const* d_in, const int* in_sizes, int n_in,
                              void* d_out, int out_size, void* d_ws, size_t ws_size,
                              hipStream_t stream) {
  const float* Q      = (const float*)d_in[0];
  const float* Z      = (const float*)d_in[1];
  const float* mask   = (const float*)d_in[2];
  const float* scorer = (const float*)d_in[3];
  const float* Wu = (const float*)d_in[4];
  const float* Uu = (const float*)d_in[5];
  const float* bu = (const float*)d_in[6];
  const float* Wr = (const float*)d_in[7];
  const float* Ur = (const float*)d_in[8];
  const float* br = (const float*)d_in[9];
  const float* Wh = (const float*)d_in[10];
  const float* Uh = (const float*)d_in[11];
  const float* bh = (const float*)d_in[12];
  float* out = (float*)d_out;
  const int N = in_sizes[2];  // mask length == number of nodes

  // workspace carve-up (all 256B-aligned)
  char* w = (char*)d_ws;
  unsigned* ctl  = (unsigned*)w;                              // 4KB
  unsigned* keys = (unsigned*)(w + 4096);
  size_t keysB   = (((size_t)N * 4) + 255) & ~(size_t)255;
  unsigned* candK  = (unsigned*)(w + 4096 + keysB);           // 8KB
  unsigned* candI  = candK + 2048;                            // 8KB
  float*    tval   = (float*)(candI + 2048);                  // 4KB
  unsigned* topIdx = (unsigned*)(tval + 1024);                // 4KB
  float* Zt   = (float*)(topIdx + 1024);                      // 1MB
  float* updW = Zt + (size_t)COLS * ROWS;                     // 1MB
  float* RQ   = updW + (size_t)ROWS * COLS;                   // 1MB

  k_init<<<1, 256, 0, stream>>>(scorer, ctl);
  k_scores<<<4096, 256, 0, stream>>>(Z, mask, scorer, ctl, keys, N);
  for (int p = 0; p < 4; ++p) {
    k_hist<<<512, 256, 0, stream>>>(keys, ctl, N, p);
    k_select<<<1, 1, 0, stream>>>(ctl, p);
  }
  k_compact<<<512, 256, 0, stream>>>(keys, ctl, candK, candI, N);
  k_sortk<<<1, 1024, 0, stream>>>(candK, candI, ctl, tval, topIdx);
  k_buildz<<<COLS, 64, 0, stream>>>(Z, tval, topIdx, Zt);
  k_gates_ur<<<COLS / 16, 512, 0, stream>>>(Wu, Uu, bu, Wr, Ur, br, Q, Zt, updW, RQ);
  k_gate_h<<<COLS / 16, 512, 0, stream>>>(Wh, Uh, bh, Q, Zt, RQ, updW, out);
}